// GAT_66529043415293
// MI455X (gfx1250) — compile-verified
//
#include <hip/hip_runtime.h>
#include <math.h>

typedef float v2f __attribute__((ext_vector_type(2)));
typedef float v8f __attribute__((ext_vector_type(8)));

#define IN_C 128
#define OUT_C 64
#define NEG_SLOPE 0.2f
#define GAT_EPS 1e-16f

// ---------------------------------------------------------------------------
// Kernel 1: h = x @ W via V_WMMA_F32_16X16X4_F32 (fp32-exact matrix pipe).
// Block = 128 threads = 4 waves. Wave w computes the 16x16 tile
// (rows blockIdx*16 .. +15) x (cols w*16 .. +15), accumulating K=128 in steps
// of 4. EXEC must stay all-ones for WMMA: reads are clamped, only the final
// stores are predicated for a ragged tail.
// ---------------------------------------------------------------------------
__global__ void gat_gemm_wmma(const float* __restrict__ x,
                              const float* __restrict__ W,
                              float* __restrict__ h, int n) {
  const int wave = threadIdx.x >> 5;
  const int lane = threadIdx.x & 31;
  const int lr   = lane & 15;   // M (for A) or N (for B/C/D)
  const int hi   = lane >> 4;   // K-half selector
  const long row0 = (long)blockIdx.x * 16;
  const int  col0 = wave * 16;

  long arow = row0 + lr;
  if (arow > n - 1) arow = n - 1;          // clamp reads; keep EXEC full
  const float* xr = x + arow * IN_C;
  const int col = col0 + lr;

  v8f acc = {};
  #pragma unroll 4
  for (int k0 = 0; k0 < IN_C; k0 += 4) {
    const int ka = k0 + 2 * hi;
    v2f a;  a.x = xr[ka];                      a.y = xr[ka + 1];
    v2f bm; bm.x = W[(long)ka * OUT_C + col];  bm.y = W[(long)(ka + 1) * OUT_C + col];
    acc = __builtin_amdgcn_wmma_f32_16x16x4_f32(false, a, false, bm,
                                                (short)0, acc, false, false);
  }

  // C/D layout: VGPR v -> row M = v (lanes 0-15) / v+8 (lanes 16-31), N = lr.
  float* hp = h + row0 * OUT_C + col0 + lr;
  #pragma unroll
  for (int v = 0; v < 8; ++v) {
    const int m = v + 8 * hi;
    if (row0 + m < n) hp[(long)m * OUT_C] = acc[v];
  }
}

// ---------------------------------------------------------------------------
// Kernel 2: per-node attention scores + init of segment buffers.
// ---------------------------------------------------------------------------
__global__ void gat_scores(const float* __restrict__ h,
                           const float* __restrict__ a_src,
                           const float* __restrict__ a_dst,
                           float* __restrict__ s_src, float* __restrict__ s_dst,
                           float* __restrict__ seg_max, float* __restrict__ denom,
                           int n) {
  const int i = blockIdx.x * blockDim.x + threadIdx.x;
  if (i >= n) return;
  const float4* hp  = (const float4*)(h + (long)i * OUT_C);
  const float4* asp = (const float4*)a_src;
  const float4* adp = (const float4*)a_dst;
  float ss = 0.f, sd = 0.f;
  #pragma unroll
  for (int c = 0; c < OUT_C / 4; ++c) {
    const float4 hv = hp[c], av = asp[c], dv = adp[c];
    ss += hv.x * av.x + hv.y * av.y + hv.z * av.z + hv.w * av.w;
    sd += hv.x * dv.x + hv.y * dv.y + hv.z * dv.z + hv.w * dv.w;
  }
  s_src[i] = ss;
  s_dst[i] = sd;
  seg_max[i] = -INFINITY;
  denom[i] = 0.f;
}

__global__ void gat_zero(float* __restrict__ p, long n) {
  const long i = (long)blockIdx.x * blockDim.x + threadIdx.x;
  if (i < n) p[i] = 0.f;
}

__device__ __forceinline__ void edge_pair(const long long* __restrict__ ei,
                                          int i, int E, int& src, int& dst) {
  if (i < E) { src = (int)ei[i]; dst = (int)ei[(long)E + i]; }
  else       { src = dst = i - E; }   // self loop
}

__device__ __forceinline__ float leaky(float v) {
  return v >= 0.f ? v : NEG_SLOPE * v;
}

// ---------------------------------------------------------------------------
// Kernel 3: segment max of leaky-relu logits (float atomic max -> L2).
// ---------------------------------------------------------------------------
__global__ void gat_edge_max(const long long* __restrict__ ei,
                             const float* __restrict__ s_src,
                             const float* __restrict__ s_dst,
                             float* __restrict__ seg_max, int E, int n) {
  const int i = blockIdx.x * blockDim.x + threadIdx.x;
  if (i >= E + n) return;
  int src, dst;
  edge_pair(ei, i, E, src, dst);
  atomicMax(&seg_max[dst], leaky(s_src[src] + s_dst[dst]));
}

// ---------------------------------------------------------------------------
// Kernel 4: segment sum of exp(logit - max).
// ---------------------------------------------------------------------------
__global__ void gat_edge_expsum(const long long* __restrict__ ei,
                                const float* __restrict__ s_src,
                                const float* __restrict__ s_dst,
                                const float* __restrict__ seg_max,
                                float* __restrict__ denom, int E, int n) {
  const int i = blockIdx.x * blockDim.x + threadIdx.x;
  if (i >= E + n) return;
  int src, dst;
  edge_pair(ei, i, E, src, dst);
  const float l = leaky(s_src[src] + s_dst[dst]);
  atomicAdd(&denom[dst], expf(l - seg_max[dst]));
}

// ---------------------------------------------------------------------------
// Kernel 5: one wave per edge; each lane handles 2 channels of the
// coef * h[src] scatter-add into out[dst]. h/out live in L2.
// ---------------------------------------------------------------------------
__global__ void gat_scatter(const long long* __restrict__ ei,
                            const float* __restrict__ h,
                            const float* __restrict__ s_src,
                            const float* __restrict__ s_dst,
                            const float* __restrict__ seg_max,
                            const float* __restrict__ denom,
                            float* __restrict__ out, int E, int n) {
  const int lane = threadIdx.x & 31;
  const int e = blockIdx.x * (blockDim.x >> 5) + (threadIdx.x >> 5);
  if (e >= E + n) return;
  int src, dst;
  edge_pair(ei, e, E, src, dst);
  const float l = leaky(s_src[src] + s_dst[dst]);
  const float coef = expf(l - seg_max[dst]) / (denom[dst] + GAT_EPS);
  const float2 hv = ((const float2*)(h + (long)src * OUT_C))[lane];
  float* op = out + (long)dst * OUT_C + 2 * lane;
  atomicAdd(op,     coef * hv.x);
  atomicAdd(op + 1, coef * hv.y);
}

// ---------------------------------------------------------------------------
// Kernel 6: out = elu(out + b)
// ---------------------------------------------------------------------------
__global__ void gat_finish(float* __restrict__ out, const float* __restrict__ b,
                           long total) {
  const long i = (long)blockIdx.x * blockDim.x + threadIdx.x;
  if (i >= total) return;
  const float v = out[i] + b[i & (OUT_C - 1)];
  out[i] = v > 0.f ? v : expm1f(v);
}

extern "C" void kernel_launch(void* const* d_in, const int* in_sizes, int n_in,
                              void* d_out, int out_size, void* d_ws, size_t ws_size,
                              hipStream_t stream) {
  const float*     x     = (const float*)d_in[0];
  const long long* ei    = (const long long*)d_in[1];   // int64 [2, E]
  const float*     W     = (const float*)d_in[2];
  const float*     a_src = (const float*)d_in[3];
  const float*     a_dst = (const float*)d_in[4];
  const float*     bias  = (const float*)d_in[5];
  const int n = in_sizes[0] / IN_C;
  const int E = in_sizes[1] / 2;

  // workspace layout
  float* h       = (float*)d_ws;                 // n * 64
  float* s_src   = h + (size_t)n * OUT_C;        // n
  float* s_dst   = s_src + n;                    // n
  float* seg_max = s_dst + n;                    // n
  float* denom   = seg_max + n;                  // n
  float* out     = (float*)d_out;

  const int  total = E + n;                      // edges + self loops
  const long on    = (long)n * OUT_C;

  gat_gemm_wmma<<<(n + 15) / 16, 128, 0, stream>>>(x, W, h, n);
  gat_scores<<<(n + 255) / 256, 256, 0, stream>>>(h, a_src, a_dst,
                                                  s_src, s_dst, seg_max, denom, n);
  gat_zero<<<(int)((on + 255) / 256), 256, 0, stream>>>(out, on);
  gat_edge_max<<<(total + 255) / 256, 256, 0, stream>>>(ei, s_src, s_dst, seg_max, E, n);
  gat_edge_expsum<<<(total + 255) / 256, 256, 0, stream>>>(ei, s_src, s_dst,
                                                           seg_max, denom, E, n);
  gat_scatter<<<(total + 7) / 8, 256, 0, stream>>>(ei, h, s_src, s_dst,
                                                   seg_max, denom, out, E, n);
  gat_finish<<<(int)((on + 255) / 256), 256, 0, stream>>>(out, bias, on);
}